// GCNEncoder_36636071035639
// MI455X (gfx1250) — compile-verified
//
#include <hip/hip_runtime.h>
#include <hip/hip_bf16.h>

typedef __attribute__((ext_vector_type(2))) float v2f;
typedef __attribute__((ext_vector_type(8))) float v8f;

// ---------------------------------------------------------------------------
// FP32 WMMA GEMM:  C[M,N] = op(A[M,K]) * B[K,N]   (op = identity or ReLU,
// compile-time). One wave computes a 16x64 strip of C via V_WMMA_F32_16X16X4_F32.
// Layouts per CDNA5 ISA 7.12.2:
//   A 16x4 : lane l<16 -> {A[m0+l][k], A[m0+l][k+1]}, lanes 16..31 -> K+2,K+3
//   B 4x16 : lane l<16 -> {B[k][n0+l], B[k+1][n0+l]}, lanes 16..31 -> K+2,K+3
//   C 16x16: lane l<16 -> vgpr v = C[m0+v][n0+l],    lanes 16..31 -> M+8
// ---------------------------------------------------------------------------
template <bool RELU>
__global__ __launch_bounds__(256) void gemm_wmma_f32(
    const float* __restrict__ A, const float* __restrict__ B,
    float* __restrict__ C, int M, int K, int N)
{
    const int lane = threadIdx.x & 31;
    const int wave = threadIdx.x >> 5;
    const int ntiles_n = N >> 6;                 // 64-wide column tiles
    const int total = (M >> 4) * ntiles_n;
    const int tile = blockIdx.x * 8 + wave;      // wave-uniform
    if (tile >= total) return;                   // whole wave exits -> EXEC all-1s below

    const int mt = tile / ntiles_n;
    const int nt = tile - mt * ntiles_n;
    const int m0 = mt << 4;
    const int n0 = nt << 6;
    const int half = lane >> 4;                  // 0: K,K+1   1: K+2,K+3
    const int l = lane & 15;

    v8f acc0 = {}, acc1 = {}, acc2 = {}, acc3 = {};

    const float* __restrict__ arow = A + (size_t)(m0 + l) * K + 2 * half;
    const float* __restrict__ bcol = B + (size_t)(2 * half) * N + n0 + l;

    #pragma unroll 4
    for (int k = 0; k < K; k += 4) {
        float2 av = *(const float2*)(arow + k);
        v2f a;
        if (RELU) { a.x = fmaxf(av.x, 0.0f); a.y = fmaxf(av.y, 0.0f); }
        else      { a.x = av.x;              a.y = av.y;              }

        const float* bp = bcol + (size_t)k * N;
        v2f b0, b1, b2, b3;
        b0.x = bp[0];  b0.y = bp[N];
        b1.x = bp[16]; b1.y = bp[N + 16];
        b2.x = bp[32]; b2.y = bp[N + 32];
        b3.x = bp[48]; b3.y = bp[N + 48];

        acc0 = __builtin_amdgcn_wmma_f32_16x16x4_f32(false, a, false, b0, (short)0, acc0, false, false);
        acc1 = __builtin_amdgcn_wmma_f32_16x16x4_f32(false, a, false, b1, (short)0, acc1, false, false);
        acc2 = __builtin_amdgcn_wmma_f32_16x16x4_f32(false, a, false, b2, (short)0, acc2, false, false);
        acc3 = __builtin_amdgcn_wmma_f32_16x16x4_f32(false, a, false, b3, (short)0, acc3, false, false);
    }

    float* crow = C + (size_t)(m0 + 8 * half) * N + n0 + l;
    #pragma unroll
    for (int v = 0; v < 8; ++v) {
        crow[(size_t)v * N +  0] = acc0[v];
        crow[(size_t)v * N + 16] = acc1[v];
        crow[(size_t)v * N + 32] = acc2[v];
        crow[(size_t)v * N + 48] = acc3[v];
    }
}

// ---------------------------------------------------------------------------
// Degree / normalization kernels
// ---------------------------------------------------------------------------
__global__ void k_set_ones(float* __restrict__ deg, int n) {
    int i = blockIdx.x * blockDim.x + threadIdx.x;
    if (i < n) deg[i] = 1.0f;                    // self-loop contribution
}

__global__ void k_degree(const long long* __restrict__ col, float* __restrict__ deg, int E) {
    int e = blockIdx.x * blockDim.x + threadIdx.x;
    if (e < E) atomicAdd(&deg[(int)col[e]], 1.0f);
}

__global__ void k_inv_sqrt(float* __restrict__ deg, int n) {
    int i = blockIdx.x * blockDim.x + threadIdx.x;
    if (i < n) {
        float d = deg[i];
        deg[i] = (d > 0.0f) ? rsqrtf(d) : 0.0f;  // deg>=1 always (self-loop), guard anyway
    }
}

// ---------------------------------------------------------------------------
// Aggregation init: out[i][c] = h[i][c]*dinv[i]^2 + bias[c]
// (fuses self-loop edge + bias; writes every element -> no zero pass)
// Vectorized float4; C is a multiple of 4.
// ---------------------------------------------------------------------------
__global__ void k_init_agg(const float* __restrict__ h, const float* __restrict__ dinv,
                           const float* __restrict__ bias, float* __restrict__ out,
                           int n, int C)
{
    const int C4 = C >> 2;
    long long i4 = (long long)blockIdx.x * blockDim.x + threadIdx.x;
    long long total4 = (long long)n * C4;
    if (i4 >= total4) return;
    int node = (int)(i4 / C4);
    int c4   = (int)(i4 - (long long)node * C4);
    float d  = dinv[node];
    float d2 = d * d;
    float4 hv = ((const float4*)h)[i4];
    float4 bv = ((const float4*)bias)[c4];
    float4 o;
    o.x = hv.x * d2 + bv.x;
    o.y = hv.y * d2 + bv.y;
    o.z = hv.z * d2 + bv.z;
    o.w = hv.w * d2 + bv.w;
    ((float4*)out)[i4] = o;
}

// ---------------------------------------------------------------------------
// Edge aggregation: out[col[e]][:] += h[row[e]][:] * dinv[row]*dinv[col]
// One wave per edge; lane covers a contiguous float4 chunk (global_load_b128
// gathers, per-float global_atomic_add_f32 scatter). Working set (~115 MB)
// is resident in the 192 MB L2, so this phase runs at L2 rates.
// ---------------------------------------------------------------------------
template <int C>
__global__ __launch_bounds__(256) void k_edge_agg(
    const long long* __restrict__ row, const long long* __restrict__ col,
    const float* __restrict__ dinv, const float* __restrict__ h,
    float* __restrict__ out, int E)
{
    int e = (int)((blockIdx.x * 256u + threadIdx.x) >> 5);
    int lane = threadIdx.x & 31;
    if (e >= E) return;

    int r = (int)row[e];
    int c = (int)col[e];
    float norm = dinv[r] * dinv[c];

    const float4* __restrict__ hp = (const float4*)(h + (size_t)r * C);
    float* __restrict__ op = out + (size_t)c * C;

    #pragma unroll
    for (int j4 = lane; j4 < (C >> 2); j4 += 32) {
        float4 v = hp[j4];
        int j = j4 << 2;
        atomicAdd(&op[j + 0], v.x * norm);
        atomicAdd(&op[j + 1], v.y * norm);
        atomicAdd(&op[j + 2], v.z * norm);
        atomicAdd(&op[j + 3], v.w * norm);
    }
}

// ---------------------------------------------------------------------------
// Launch: deg -> dinv | GEMM1 | init+edge agg (256ch) | GEMM2(fused ReLU) |
//         init+edge agg (128ch) into d_out
// ---------------------------------------------------------------------------
extern "C" void kernel_launch(void* const* d_in, const int* in_sizes, int n_in,
                              void* d_out, int out_size, void* d_ws, size_t ws_size,
                              hipStream_t stream)
{
    const float*     x    = (const float*)d_in[0];
    const long long* edge = (const long long*)d_in[1];   // int64 in reference
    const float*     W1   = (const float*)d_in[2];
    const float*     b1   = (const float*)d_in[3];
    const float*     W2   = (const float*)d_in[4];
    const float*     b2   = (const float*)d_in[5];
    float*           out  = (float*)d_out;

    const int IN_C = 256, HID_C = 256, OUT_C = 128;
    const int N = in_sizes[0] / IN_C;     // 50000
    const int E = in_sizes[1] / 2;        // 800000

    const long long* erow = edge;         // [2,E] row-major: row then col
    const long long* ecol = edge + E;

    // Workspace layout (floats): dinv | h1 | agg1 | h2
    float* ws   = (float*)d_ws;
    float* dinv = ws;
    float* h1   = dinv + N;
    float* agg1 = h1 + (size_t)N * HID_C;
    float* h2   = agg1 + (size_t)N * HID_C;

    // --- degrees and d^{-1/2} (recomputed every call: deterministic) ---
    k_set_ones<<<(N + 255) / 256, 256, 0, stream>>>(dinv, N);
    k_degree  <<<(E + 255) / 256, 256, 0, stream>>>(ecol, dinv, E);
    k_inv_sqrt<<<(N + 255) / 256, 256, 0, stream>>>(dinv, N);

    // --- layer 1: h1 = x @ W1 ---
    {
        int tiles = (N / 16) * (HID_C / 64);
        gemm_wmma_f32<false><<<(tiles + 7) / 8, 256, 0, stream>>>(x, W1, h1, N, IN_C, HID_C);
    }
    // agg1 = self-loop + bias, then scatter-add over edges
    {
        long long tot4 = (long long)N * (HID_C / 4);
        k_init_agg<<<(unsigned)((tot4 + 255) / 256), 256, 0, stream>>>(h1, dinv, b1, agg1, N, HID_C);
        k_edge_agg<256><<<(E + 7) / 8, 256, 0, stream>>>(erow, ecol, dinv, h1, agg1, E);
    }

    // --- layer 2: h2 = relu(agg1) @ W2   (ReLU fused into A-load) ---
    {
        int tiles = (N / 16) * (OUT_C / 64);
        gemm_wmma_f32<true><<<(tiles + 7) / 8, 256, 0, stream>>>(agg1, W2, h2, N, HID_C, OUT_C);
    }
    // out = self-loop + bias, then scatter-add over edges
    {
        long long tot4 = (long long)N * (OUT_C / 4);
        k_init_agg<<<(unsigned)((tot4 + 255) / 256), 256, 0, stream>>>(h2, dinv, b2, out, N, OUT_C);
        k_edge_agg<128><<<(E + 7) / 8, 256, 0, stream>>>(erow, ecol, dinv, h2, out, E);
    }
}